// MultiFrequencyChannelAttention_292057776143
// MI455X (gfx1250) — compile-verified
//
#include <hip/hip_runtime.h>
#include <hip/hip_bf16.h>

typedef float v2f __attribute__((ext_vector_type(2)));
typedef float v4f __attribute__((ext_vector_type(4)));
typedef float v8f __attribute__((ext_vector_type(8)));

#define DCT_HW   7
#define NFREQ    16
#define HW       56
#define PIX      (HW * HW)        // 3136
#define PIX4     (PIX / 4)        // 784
#define BATCH    32
#define CH       256
#define BC       (BATCH * CH)     // 8192
#define RED      16               // C / 16

// ---------------------------------------------------------------------------
// Kernel 1: one block per (b,c). Load 56x56 tile to LDS (b128 coalesced),
// 8x8 uniform pooling to 7x7, then per-frequency sum/max/min stats.
// ---------------------------------------------------------------------------
__global__ __launch_bounds__(256)
void pool_feat_kernel(const float* __restrict__ x,
                      const float* __restrict__ dctf,   // [16,49]
                      float* __restrict__ feat_avg,
                      float* __restrict__ feat_max,
                      float* __restrict__ feat_min) {
  __shared__ float tile[PIX];
  __shared__ float xp[49];
  __shared__ float s_sum[NFREQ], s_max[NFREQ], s_min[NFREQ];

  const int bc = blockIdx.x;
  const v4f* __restrict__ xv =
      reinterpret_cast<const v4f*>(x + (size_t)bc * PIX);
  v4f* tv = reinterpret_cast<v4f*>(tile);
  for (int i = threadIdx.x; i < PIX4; i += 256) tv[i] = xv[i];
  __syncthreads();

  const int t = threadIdx.x;
  if (t < 49) {                       // pooled cell (i,j): mean of 8x8 block
    const int ci = t / 7, cj = t % 7;
    const int base = ci * 8 * HW + cj * 8;
    float s = 0.f;
#pragma unroll
    for (int r = 0; r < 8; ++r) {
      const int rb = base + r * HW;
#pragma unroll
      for (int q = 0; q < 8; ++q) s += tile[rb + q];
    }
    xp[t] = s * (1.0f / 64.0f);
  }
  __syncthreads();

  if (t < NFREQ) {                    // per-frequency stats over 49 cells
    float sacc = 0.f, mx = -__builtin_inff(), mn = __builtin_inff();
    const float* __restrict__ f = dctf + t * 49;
    for (int c = 0; c < 49; ++c) {
      const float v = xp[c] * f[c];
      sacc += v;
      mx = fmaxf(mx, v);
      mn = fminf(mn, v);
    }
    s_sum[t] = sacc; s_max[t] = mx; s_min[t] = mn;
  }
  __syncthreads();

  if (t == 0) {                       // mean over the 16 frequencies
    float a = 0.f, mxs = 0.f, mns = 0.f;
#pragma unroll
    for (int f = 0; f < NFREQ; ++f) { a += s_sum[f]; mxs += s_max[f]; mns += s_min[f]; }
    feat_avg[bc] = a * (1.0f / (49.0f * (float)NFREQ));
    feat_max[bc] = mxs * (1.0f / (float)NFREQ);
    feat_min[bc] = mns * (1.0f / (float)NFREQ);
  }
}

// ---------------------------------------------------------------------------
// Kernel 2: SE MLP via V_WMMA_F32_16X16X4_F32, single block of 8 wave32 waves.
// Stage A: y_s = relu(feat_s @ w1^T)  -- 6 tile jobs (3 feats x 2 M-tiles),
//          K=256 in steps of 4 (64 wmma per tile).
// Stage B: z = (y_avg+y_max+y_min) @ w2^T  -- 32 tiles x 4 k-steps,
//          att = sigmoid(z).
// All WMMA-adjacent branches are wave-uniform => EXEC all-1s (ISA 7.12 req).
// ---------------------------------------------------------------------------
__global__ __launch_bounds__(256)
void mlp_att_kernel(const float* __restrict__ w1,     // [16,256]
                    const float* __restrict__ w2,     // [256,16]
                    const float* __restrict__ feats,  // 3 x [32,256] contiguous
                    float* __restrict__ att) {        // [32,256]
  __shared__ float y[3 * BATCH * RED];   // y[s][b][n]
  __shared__ float ysum[BATCH * RED];

  const int tid  = threadIdx.x;
  const int wave = tid >> 5;
  const int lane = tid & 31;
  const int g    = lane >> 4;   // half-wave group (selects K pair / M block)
  const int lr   = lane & 15;

  // ---- Stage A: first FC + relu --------------------------------------
  if (wave < 6) {
    const int s     = wave >> 1;
    const int mtile = wave & 1;
    const float* __restrict__ feat = feats + s * (BATCH * CH);
    const int row = mtile * 16 + lr;            // A: M = batch row
    v8f acc = {};
    for (int k0 = 0; k0 < CH; k0 += 4) {
      // A 16x4 fp32 layout: lanes 0-15 -> K=k0,k0+1 ; lanes 16-31 -> K=k0+2,k0+3
      v2f a = *reinterpret_cast<const v2f*>(feat + row * CH + k0 + 2 * g);
      // B 4x16: N = lr, K mirrored across half-waves; B[k][n] = w1[n][k]
      v2f b = *reinterpret_cast<const v2f*>(w1 + lr * CH + k0 + 2 * g);
      acc = __builtin_amdgcn_wmma_f32_16x16x4_f32(
          false, a, false, b, (short)0, acc, false, false);
    }
#pragma unroll
    for (int r = 0; r < 8; ++r) {               // D layout: M = r + 8*g, N = lr
      const int brow = mtile * 16 + r + 8 * g;
      const float v = acc[r];
      y[s * (BATCH * RED) + brow * RED + lr] = v > 0.f ? v : 0.f;
    }
  }
  __syncthreads();

  // ysum = y_avg + y_max + y_min  (second FC is linear, fold the 3 paths)
  for (int e = tid; e < BATCH * RED; e += 256)
    ysum[e] = y[e] + y[BATCH * RED + e] + y[2 * BATCH * RED + e];
  __syncthreads();

  // ---- Stage B: second FC + sigmoid ----------------------------------
  for (int tI = wave; tI < 32; tI += 8) {
    const int mtile = tI >> 4;                  // 0..1   (batch tile)
    const int ntile = tI & 15;                  // 0..15  (channel tile)
    const int arow = mtile * 16 + lr;
    const int ccol = ntile * 16 + lr;
    v8f acc = {};
#pragma unroll
    for (int ks = 0; ks < 4; ++ks) {
      const int k = ks * 4 + 2 * g;
      v2f a = *reinterpret_cast<const v2f*>(&ysum[arow * RED + k]);
      v2f b = *reinterpret_cast<const v2f*>(w2 + ccol * RED + k);  // w2[c][k]
      acc = __builtin_amdgcn_wmma_f32_16x16x4_f32(
          false, a, false, b, (short)0, acc, false, false);
    }
#pragma unroll
    for (int r = 0; r < 8; ++r) {
      const int brow = mtile * 16 + r + 8 * g;
      const float z = acc[r];
      att[brow * CH + ccol] = 1.0f / (1.0f + __expf(-z));
    }
  }
}

// ---------------------------------------------------------------------------
// Kernel 3: out = x * att. One float4 per thread. x should still be L2-
// resident from kernel 1 (102.8 MB < 192 MB L2); store nontemporal so the
// output stream does not evict it / pollute L2.
// ---------------------------------------------------------------------------
__global__ __launch_bounds__(256)
void scale_kernel(const float* __restrict__ x,
                  const float* __restrict__ att,
                  float* __restrict__ out, int n4) {
  const int i = blockIdx.x * 256 + threadIdx.x;
  if (i >= n4) return;
  const int bc = i / PIX4;                      // 784 float4 per (b,c)
  const float a = att[bc];
  v4f v = reinterpret_cast<const v4f*>(x)[i];
  v *= a;
  __builtin_nontemporal_store(v, reinterpret_cast<v4f*>(out) + i);
}

// ---------------------------------------------------------------------------
extern "C" void kernel_launch(void* const* d_in, const int* in_sizes, int n_in,
                              void* d_out, int out_size, void* d_ws, size_t ws_size,
                              hipStream_t stream) {
  const float* x    = (const float*)d_in[0];   // [32,256,56,56]
  const float* w1   = (const float*)d_in[1];   // [16,256]
  const float* w2   = (const float*)d_in[2];   // [256,16]
  const float* dctf = (const float*)d_in[3];   // [16,7,7]
  float* out = (float*)d_out;

  float* ws       = (float*)d_ws;
  float* feat_avg = ws;                 // [8192]
  float* feat_max = ws + BC;            // [8192]
  float* feat_min = ws + 2 * BC;        // [8192]
  float* att      = ws + 3 * BC;        // [8192]   (total 128 KB scratch)

  pool_feat_kernel<<<BC, 256, 0, stream>>>(x, dctf, feat_avg, feat_max, feat_min);
  mlp_att_kernel<<<1, 256, 0, stream>>>(w1, w2, ws, att);

  const int n4 = BC * PIX4;             // 6,422,528 float4
  scale_kernel<<<(n4 + 255) / 256, 256, 0, stream>>>(x, att, out, n4);
}